// CapsuleLayer_17592186044964
// MI455X (gfx1250) — compile-verified
//
#include <hip/hip_runtime.h>
#include <hip/hip_bf16.h>

typedef __attribute__((ext_vector_type(2))) float v2f;
typedef __attribute__((ext_vector_type(8))) float v8f;
typedef __attribute__((ext_vector_type(2))) int   v2i;

#define B_ 64
#define I_ 1152
#define D_ 16
#define J_ 32
#define E_ 32
#define EPS_ 1e-7f

// gfx1250 async global->LDS path, guarded so the source always compiles.
#if defined(__has_builtin)
#if __has_builtin(__builtin_amdgcn_global_load_async_to_lds_b64) && \
    __has_builtin(__builtin_amdgcn_s_wait_asynccnt)
#define USE_ASYNC_LDS 1
#endif
#endif
#ifndef USE_ASYNC_LDS
#define USE_ASYNC_LDS 0
#endif

// builtin signature (from compiler diagnostic): (as1 v2i* src, as3 v2i* dst, imm, imm)
typedef __attribute__((address_space(1))) v2i* gasync_t;
typedef __attribute__((address_space(3))) v2i* lasync_t;

// ---------------- zero the routing logits ----------------
__global__ void caps_zero(float* __restrict__ p, int n) {
  int t = blockIdx.x * blockDim.x + threadIdx.x;
  if (t < n) p[t] = 0.f;
}

// ---------------- u_hat GEMM: fp32 WMMA, bf16 output ----------------
// Per input capsule i: A = inputs[:,i,:] (64x16), B = kernel[i] (16 x 1024),
// tiled into 16x16 output tiles, one wave per tile, 4x V_WMMA_F32_16X16X4_F32.
__global__ void caps_uhat_wmma(const float* __restrict__ inp,
                               const float* __restrict__ ker,
                               __hip_bfloat16* __restrict__ uhat) {
  const int i    = blockIdx.x;
  const int lane = threadIdx.x & 31;
  const int tid  = blockIdx.y * 8 + (threadIdx.x >> 5); // 0..255 tile id
  const int mt   = tid >> 6;          // 4 M-tiles (batch rows)
  const int nt   = tid & 63;          // 64 N-tiles over (j,e)
  const int j    = nt >> 1;
  const int e0   = (nt & 1) << 4;
  const int m    = lane & 15;         // A: M row / B,C: N column
  const int h    = lane >> 4;         // lane half

  const float* Abase = inp + ((size_t)(mt * 16 + m) * I_ + i) * D_;
  const float* Bbase = ker + (((size_t)i * J_ + j) * D_) * E_ + e0 + m;

  v8f acc = {};
#pragma unroll
  for (int kb = 0; kb < 16; kb += 4) {
    // A 16x4 frag: VGPR v, half h -> K = kb + v + 2h (contiguous pair -> b64 load)
    v2f a = *(const v2f*)(Abase + kb + 2 * h);
    // B 4x16 frag: same K mapping, column = lane&15
    v2f b;
    b[0] = Bbase[(size_t)(kb + 2 * h)     * E_];
    b[1] = Bbase[(size_t)(kb + 2 * h + 1) * E_];
    acc = __builtin_amdgcn_wmma_f32_16x16x4_f32(false, a, false, b,
                                                (short)0, acc, false, false);
  }

  // C/D layout: VGPR r, half h -> M = r + 8h ; N = lane&15
  __hip_bfloat16* out = uhat + (((size_t)i * J_ + j) * E_) + e0 + m;
  const size_t bstride = (size_t)I_ * J_ * E_;
#pragma unroll
  for (int r = 0; r < 8; ++r) {
    int br = mt * 16 + r + 8 * h;
    out[(size_t)br * bstride] = __float2bfloat16(acc[r]);
  }
}

// ---------------- softmax over j per input capsule i ----------------
__global__ void caps_softmax(const float* __restrict__ bias,
                             float* __restrict__ c) {
  int i = blockIdx.x * 4 + (threadIdx.x >> 5);
  int j = threadIdx.x & 31;
  float v = bias[i * J_ + j];
  float mx = v;
#pragma unroll
  for (int s = 16; s >= 1; s >>= 1) mx = fmaxf(mx, __shfl_xor(mx, s, 32));
  float ex = expf(v - mx);
  float sm = ex;
#pragma unroll
  for (int s = 16; s >= 1; s >>= 1) sm += __shfl_xor(sm, s, 32);
  c[i * J_ + j] = ex / sm;
}

// ---------------- s = sum_i c*u_hat ; out = squash(s) ----------------
// one block per (b,j). Double-buffered async global->LDS staging of u_hat
// (64 i-rows = 4KB per stage), consumption overlapped with next-stage DMA.
#define STAGE_ROWS 64
__global__ void caps_route_squash(const __hip_bfloat16* __restrict__ uhat,
                                  const float* __restrict__ c,
                                  float* __restrict__ out) {
  __shared__ __hip_bfloat16 tile[2][STAGE_ROWS][E_]; // 2 x 4KB
  __shared__ float part[256];
  const int b = blockIdx.x, j = blockIdx.y;
  const int t = threadIdx.x;
  const int e = t & 31, chunk = t >> 5;
  const __hip_bfloat16* ubase = uhat + (((size_t)b * I_) * J_ + j) * E_;
  const size_t istride = (size_t)J_ * E_; // elements between i-rows
  float acc = 0.f;

#if USE_ASYNC_LDS
  // chunk id ch in [0,512): row = ch>>3, qword = ch&7 ; 2 chunks per thread
  const int ch0 = t * 2;
  const int row0 = ch0 >> 3, qw0 = ch0 & 7;
  const int ch1 = ch0 + 1;
  const int row1 = ch1 >> 3, qw1 = ch1 & 7;
  const int nstages = I_ / STAGE_ROWS; // 18

  // issue stage 0
  {
    gasync_t g0 = (gasync_t)(uintptr_t)(ubase + (size_t)row0 * istride + qw0 * 4);
    lasync_t l0 = (lasync_t)(uint32_t)(uintptr_t)&tile[0][row0][qw0 * 4];
    __builtin_amdgcn_global_load_async_to_lds_b64(g0, l0, 0, 0);
    gasync_t g1 = (gasync_t)(uintptr_t)(ubase + (size_t)row1 * istride + qw1 * 4);
    lasync_t l1 = (lasync_t)(uint32_t)(uintptr_t)&tile[0][row1][qw1 * 4];
    __builtin_amdgcn_global_load_async_to_lds_b64(g1, l1, 0, 0);
  }
  for (int s = 0; s < nstages; ++s) {
    const int cur = s & 1;
    const int i0 = s * STAGE_ROWS;
    if (s + 1 < nstages) {
      const size_t inext = (size_t)(i0 + STAGE_ROWS);
      gasync_t g0 = (gasync_t)(uintptr_t)(ubase + (inext + row0) * istride + qw0 * 4);
      lasync_t l0 = (lasync_t)(uint32_t)(uintptr_t)&tile[cur ^ 1][row0][qw0 * 4];
      __builtin_amdgcn_global_load_async_to_lds_b64(g0, l0, 0, 0);
      gasync_t g1 = (gasync_t)(uintptr_t)(ubase + (inext + row1) * istride + qw1 * 4);
      lasync_t l1 = (lasync_t)(uint32_t)(uintptr_t)&tile[cur ^ 1][row1][qw1 * 4];
      __builtin_amdgcn_global_load_async_to_lds_b64(g1, l1, 0, 0);
      __builtin_amdgcn_s_wait_asynccnt(2); // stage s landed (in-order), s+1 in flight
    } else {
      __builtin_amdgcn_s_wait_asynccnt(0);
    }
    __syncthreads(); // all waves' DMA for stage s visible
#pragma unroll
    for (int r = chunk; r < STAGE_ROWS; r += 8)
      acc += c[(i0 + r) * J_ + j] * __bfloat162float(tile[cur][r][e]);
    __syncthreads(); // done reading tile[cur] before it is re-issued
  }
#else
  const __hip_bfloat16* up = ubase + e;
  for (int i = chunk; i < I_; i += 8) {
    __builtin_prefetch(up + (size_t)(i + 64) * istride, 0, 0);
    acc += c[i * J_ + j] * __bfloat162float(up[(size_t)i * istride]);
  }
#endif

  part[t] = acc;
  __syncthreads();
  if (t < 32) {
    float s = 0.f;
#pragma unroll
    for (int k = 0; k < 8; ++k) s += part[k * 32 + t];
    float s2 = s * s;
#pragma unroll
    for (int m = 16; m >= 1; m >>= 1) s2 += __shfl_xor(s2, m, 32);
    float scale = s2 / ((1.f + s2) * sqrtf(s2 + EPS_));
    out[((size_t)b * J_ + j) * E_ + t] = scale * s;
  }
}

// ---------------- bias += sum_{b,e} u_hat * outputs ----------------
// one block per i; wave w covers j = w, w+8, w+16, w+24 ; lane = e
__global__ void caps_bias_update(const __hip_bfloat16* __restrict__ uhat,
                                 const float* __restrict__ outp,
                                 float* __restrict__ bias) {
  const int i = blockIdx.x;
  const int w = threadIdx.x >> 5;
  const int e = threadIdx.x & 31;
  const size_t bstride = (size_t)I_ * J_ * E_;
#pragma unroll
  for (int jj = 0; jj < 4; ++jj) {
    const int j = w + jj * 8;
    const __hip_bfloat16* up = uhat + (((size_t)i * J_ + j) * E_) + e;
    float acc = 0.f;
    for (int b = 0; b < B_; ++b)
      acc += __bfloat162float(up[(size_t)b * bstride]) *
             outp[((size_t)b * J_ + j) * E_ + e];
#pragma unroll
    for (int m = 16; m >= 1; m >>= 1) acc += __shfl_xor(acc, m, 32);
    if (e == 0) bias[i * J_ + j] += acc;
  }
}

extern "C" void kernel_launch(void* const* d_in, const int* in_sizes, int n_in,
                              void* d_out, int out_size, void* d_ws, size_t ws_size,
                              hipStream_t stream) {
  (void)in_sizes; (void)n_in; (void)out_size; (void)ws_size;
  const float* inp = (const float*)d_in[0];   // [64,1152,16]
  const float* ker = (const float*)d_in[1];   // [1152,32,16,32]
  float* out = (float*)d_out;                 // [64,32,32]

  char* ws = (char*)d_ws;
  __hip_bfloat16* uhat = (__hip_bfloat16*)ws;                 // 151 MB, L2-resident
  size_t off = (size_t)B_ * I_ * J_ * E_ * sizeof(__hip_bfloat16);
  float* bias = (float*)(ws + off); off += (size_t)I_ * J_ * sizeof(float);
  float* cbuf = (float*)(ws + off); off += (size_t)I_ * J_ * sizeof(float);
  float* sbuf = (float*)(ws + off);

  caps_zero<<<(I_ * J_ + 255) / 256, 256, 0, stream>>>(bias, I_ * J_);
  caps_uhat_wmma<<<dim3(I_, 32), 256, 0, stream>>>(inp, ker, uhat);

  for (int r = 0; r < 3; ++r) {
    caps_softmax<<<I_ / 4, 128, 0, stream>>>(bias, cbuf);
    float* o = (r == 2) ? out : sbuf;
    caps_route_squash<<<dim3(B_, J_), 256, 0, stream>>>(uhat, cbuf, o);
    if (r < 2)
      caps_bias_update<<<I_, 256, 0, stream>>>(uhat, sbuf, bias);
  }
}